// ModelNew_3556232922104
// MI455X (gfx1250) — compile-verified
//
#include <hip/hip_runtime.h>

typedef __attribute__((ext_vector_type(2))) float v2f;
typedef __attribute__((ext_vector_type(8))) float v8f;

#define BATCH   4096
#define KDIM    4096
#define HDIM    4096
#define HCH     32                 // pass-1 H chunks
#define KSPLIT  8                  // pass-2 K chunks
#define KC      (KDIM / KSPLIT)    // 512 k per wave
#define NB      (KC / 16)          // 32 pipelined blocks of 16 k
#define TILES   (BATCH / 16)       // 256 row tiles
#define WPB     8                  // waves per block

// ---------------- Pass 1: partial column sums of W (float4 streaming) -------
__global__ void colsum_partial_kernel(const float* __restrict__ w,
                                      float* __restrict__ partial) {
  const int k4   = blockIdx.x * blockDim.x + threadIdx.x;   // 0..1023
  const int hc   = blockIdx.y;                              // 0..31
  const int rows = HDIM / HCH;                              // 128
  const float4* p = (const float4*)w + (size_t)hc * rows * (KDIM / 4) + k4;
  float4 s = make_float4(0.f, 0.f, 0.f, 0.f);
  #pragma unroll 8
  for (int h = 0; h < rows; ++h) {
    float4 v = p[(size_t)h * (KDIM / 4)];
    s.x += v.x; s.y += v.y; s.z += v.z; s.w += v.w;
  }
  ((float4*)partial)[hc * (KDIM / 4) + k4] = s;
}

// ---------------- Pass 1b: fold HCH partials into w_sum ---------------------
__global__ void colsum_reduce_kernel(const float* __restrict__ partial,
                                     float* __restrict__ wsum) {
  const int k4 = blockIdx.x * blockDim.x + threadIdx.x;     // 0..1023
  float4 s = make_float4(0.f, 0.f, 0.f, 0.f);
  #pragma unroll
  for (int hc = 0; hc < HCH; ++hc) {
    float4 v = ((const float4*)partial)[hc * (KDIM / 4) + k4];
    s.x += v.x; s.y += v.y; s.z += v.z; s.w += v.w;
  }
  ((float4*)wsum)[k4] = s;
}

// ---------------- Pass 2: row dots via V_WMMA_F32_16X16X4_F32 ---------------
// Wave = (16-row tile) x (K chunk of 512).  Software-pipelined: prefetch the
// next 16-k block (8 b64 loads) while 4 independent WMMAs consume the current.
//   A (16x4): lane L holds row (L&15), K-pair 2*(L>>4)  -> per-lane float2
//   B (4x16): lane L holds wsum[k+2*(L>>4)..+1], broadcast across columns
__global__ void rowdot_wmma_kernel(const float* __restrict__ x,
                                   const float* __restrict__ wsum,
                                   float* __restrict__ pp) {
  const int lane  = threadIdx.x & 31;
  const int wave  = threadIdx.x >> 5;                       // 0..7
  const int tile  = blockIdx.x * WPB + wave;                // 0..255
  const int kc    = blockIdx.y;                             // 0..7
  const int row   = lane & 15;
  const int khalf = (lane >> 4) << 1;                       // 0 or 2

  const float* xp = x + (size_t)(tile * 16 + row) * KDIM + kc * KC + khalf;
  const float* wp = wsum + kc * KC + khalf;

  v2f a[4], b[4];
  #pragma unroll
  for (int j = 0; j < 4; ++j) {
    a[j] = *(const v2f*)(xp + 4 * j);
    b[j] = *(const v2f*)(wp + 4 * j);
  }

  v8f c[4] = {v8f{}, v8f{}, v8f{}, v8f{}};

  for (int kb = 1; kb <= NB; ++kb) {
    v2f na[4], nb[4];
    const bool more = (kb < NB);
    if (more) {
      const int off = kb * 16;
      #pragma unroll
      for (int j = 0; j < 4; ++j) {
        na[j] = *(const v2f*)(xp + off + 4 * j);
        nb[j] = *(const v2f*)(wp + off + 4 * j);
      }
    }
    #pragma unroll
    for (int j = 0; j < 4; ++j)
      c[j] = __builtin_amdgcn_wmma_f32_16x16x4_f32(false, a[j], false, b[j],
                                                   (short)0, c[j], false, false);
    if (more) {
      #pragma unroll
      for (int j = 0; j < 4; ++j) { a[j] = na[j]; b[j] = nb[j]; }
    }
  }

  // C layout: VGPR j = rows j (lanes 0-15) / j+8 (lanes 16-31); all columns
  // identical.  Lanes 0 and 16 emit the 16 per-(tile,kc) partials.
  if ((lane & 15) == 0) {
    const int base = kc * BATCH + tile * 16 + (lane >> 4) * 8;
    #pragma unroll
    for (int j = 0; j < 8; ++j)
      pp[base + j] = c[0][j] + c[1][j] + c[2][j] + c[3][j];
  }
}

// ---------------- Pass 2b: fold KSPLIT partials into out --------------------
__global__ void out_reduce_kernel(const float* __restrict__ pp,
                                  float* __restrict__ out) {
  const int row = blockIdx.x * blockDim.x + threadIdx.x;    // 0..4095
  float s = 0.f;
  #pragma unroll
  for (int kc = 0; kc < KSPLIT; ++kc) s += pp[kc * BATCH + row];
  out[row] = s * (0.5f * 2.0f);
}

extern "C" void kernel_launch(void* const* d_in, const int* in_sizes, int n_in,
                              void* d_out, int out_size, void* d_ws, size_t ws_size,
                              hipStream_t stream) {
  const float* x = (const float*)d_in[0];   // (4096, 4096) f32
  const float* w = (const float*)d_in[1];   // (4096, 4096) f32
  float* out = (float*)d_out;               // (4096, 1)   f32

  float* partial1 = (float*)d_ws;                 // HCH   * KDIM  floats
  float* wsum     = partial1 + HCH * KDIM;        // KDIM          floats
  float* pp       = wsum + KDIM;                  // KSPLIT * BATCH floats

  colsum_partial_kernel<<<dim3(KDIM / 4 / 256, HCH), 256, 0, stream>>>(w, partial1);
  colsum_reduce_kernel<<<dim3(KDIM / 4 / 256), 256, 0, stream>>>(partial1, wsum);
  rowdot_wmma_kernel<<<dim3(TILES / WPB, KSPLIT), 256, 0, stream>>>(x, wsum, pp);
  out_reduce_kernel<<<dim3(BATCH / 256), 256, 0, stream>>>(pp, out);
}